// Block_84679575208053
// MI455X (gfx1250) — compile-verified
//
#include <hip/hip_runtime.h>
#include <math.h>
#include <stdint.h>

// ---------------------------------------------------------------------------
// Transformer block (LN -> attn -> residual -> top2 MoE adapters -> LN -> MLP)
// for MI455X / gfx1250.  All GEMMs run on V_WMMA_F32_16X16X4_F32 (exact fp32
// matrix op) so numerics match the fp32 JAX reference while using the CDNA5
// matrix pipeline.  wave32 everywhere.
//
// This revision stages GEMM B-tiles through LDS with CDNA5 async global->LDS
// DMA (GLOBAL_LOAD_ASYNC_TO_LDS_B128 / s_wait_asynccnt), double-buffered, so
// each 32x64 B-chunk is fetched once per block instead of once per wave (8x
// less L2 traffic on the dominant GEMMs).
// ---------------------------------------------------------------------------

#define USE_ASYNC_LDS 1   // 0 = plain global load + ds_store fallback

#define Bn   4
#define Tn   1024
#define Cn   1024
#define Hn   16
#define Dn   64
#define En   8
#define BOTn 64
#define FFn  4096
#define BTn  (Bn * Tn)          // 4096 rows

typedef float v2f __attribute__((ext_vector_type(2)));
typedef float v8f __attribute__((ext_vector_type(8)));

// ---------------------------------------------------------------------------
// Generic fp32 WMMA GEMM:  out = epilogue(A[M,K] @ W[K,N] + bias[N])
// Block = 256 threads = 8 waves; each wave computes a 16(M) x 64(N) strip of
// a 128x64 block tile.  B is staged in LDS:
//   chunk = 32(K) x 64(N) floats, row stride padded to 72 words so the two
//   16-lane halves of a fragment read hit disjoint bank ranges.
//   Double buffered; filled by async global->LDS b128 DMA (2 per thread).
// A-fragment (16x4 f32): lanes 0-15 hold K=k,k+1 ; lanes 16-31 hold K=k+2,k+3.
// C/D (16x16 f32): VGPR r -> row r (lanes 0-15) / row r+8 (lanes 16-31).
//
// ACT: 0=none, 1=relu, 2=exact gelu.  RES1/RES2: add residual tensors.
// GACC: out[idx] += gates[t,expert] * 0.1 * v   (MoE adapter combine)
// ---------------------------------------------------------------------------
#define GEMM_KC   32
#define GEMM_BSTR 72
#define GEMM_LDS_BYTES (2 * GEMM_KC * GEMM_BSTR * 4)

template <int ACT, bool RES1, bool RES2, bool GACC>
__global__ __launch_bounds__(256) void gemm_wmma(
    const float* __restrict__ A, const float* __restrict__ W,
    const float* __restrict__ bias,
    const float* __restrict__ res1, const float* __restrict__ res2,
    const float* __restrict__ gates, int expert,
    float* __restrict__ out, int M, int N, int K)
{
    extern __shared__ float Bs[];          // [2][GEMM_KC][GEMM_BSTR]

    const int tid  = threadIdx.x;
    const int lane = tid & 31;
    const int wave = tid >> 5;
    const int half = lane >> 4;     // 0: K pair {0,1}, 1: K pair {2,3}
    const int l15  = lane & 15;
    const int m0 = blockIdx.y * 128 + wave * 16;
    const int n0 = blockIdx.x * 64;

    const float* arow = A + (size_t)(m0 + l15) * K;
    const int nchunk = K / GEMM_KC;

    // Cooperative loader: one 32x64 B-chunk per block, 2 float4 per thread.
    auto issueChunk = [&](int kc) {
        const int k0 = kc * GEMM_KC;
        const uint32_t bufbyte = (uint32_t)((kc & 1) * (GEMM_KC * GEMM_BSTR) * 4);
#pragma unroll
        for (int p = 0; p < 2; ++p) {
            const int id = tid + p * 256;
            const int kr = id >> 4;          // 0..31  (K row in chunk)
            const int c4 = (id & 15) * 4;    // 0..60  (N col, float4 granular)
            const uint32_t laddr = bufbyte + (uint32_t)(kr * GEMM_BSTR + c4) * 4u;
#if USE_ASYNC_LDS
            const uint64_t gaddr =
                (uint64_t)(uintptr_t)(W + (size_t)(k0 + kr) * N + (n0 + c4));
            // VDST = per-lane LDS byte offset, VADDR = 64-bit global address.
            asm volatile("global_load_async_to_lds_b128 %0, %1, off"
                         :: "v"(laddr), "v"(gaddr) : "memory");
#else
            const float4 t = *(const float4*)(W + (size_t)(k0 + kr) * N + (n0 + c4));
            *(float4*)((char*)Bs + laddr) = t;
#endif
        }
    };

    issueChunk(0);

    v8f acc[4] = {};
    for (int kc = 0; kc < nchunk; ++kc) {
#if USE_ASYNC_LDS
        asm volatile("s_wait_asynccnt 0" ::: "memory");   // my chunk DMAs done
#endif
        __syncthreads();                                  // everyone's done
        if (kc + 1 < nchunk) issueChunk(kc + 1);          // prefetch other buf
        const float* buf = Bs + (kc & 1) * (GEMM_KC * GEMM_BSTR);
        const int k0 = kc * GEMM_KC;
#pragma unroll
        for (int kl = 0; kl < GEMM_KC; kl += 4) {
            const int ka = k0 + kl + half * 2;
            v2f a;
            a.x = arow[ka];
            a.y = arow[ka + 1];
            const int kb = kl + half * 2;
#pragma unroll
            for (int j = 0; j < 4; ++j) {
                const int col = j * 16 + l15;
                v2f b;
                b.x = buf[kb * GEMM_BSTR + col];
                b.y = buf[(kb + 1) * GEMM_BSTR + col];
                acc[j] = __builtin_amdgcn_wmma_f32_16x16x4_f32(
                    false, a, false, b, 0, acc[j], false, false);
            }
        }
        __syncthreads();   // all waves finished reading this buffer
    }

#pragma unroll
    for (int j = 0; j < 4; ++j) {
        const int col = n0 + j * 16 + l15;
        const float bv = bias[col];
#pragma unroll
        for (int r = 0; r < 8; ++r) {
            const int row = m0 + r + half * 8;
            float v = acc[j][r] + bv;
            if (ACT == 1) v = fmaxf(v, 0.0f);
            if (ACT == 2) v = 0.5f * v * (1.0f + erff(v * 0.70710678118654752f));
            const size_t idx = (size_t)row * N + col;
            if (RES1) v += res1[idx];
            if (RES2) v += res2[idx];
            if (GACC) {
                const float g = gates[(size_t)(row & (Tn - 1)) * En + expert];
                out[idx] += g * 0.1f * v;
            } else {
                out[idx] = v;
            }
        }
    }
}

// ---------------------------------------------------------------------------
// Causal flash attention, fp32 WMMA.  One wave per (b, h, 16-query tile);
// 4 waves per block.  Row softmax stats reduced with width-16 shfl butterflies
// (each 16x16 score tile keeps row M in VGPR index, col N in lane).
// P tile is staged through LDS to convert C-layout -> A-fragment layout.
// ---------------------------------------------------------------------------
__global__ __launch_bounds__(128) void attn_kernel(
    const float* __restrict__ q, const float* __restrict__ k,
    const float* __restrict__ v, float* __restrict__ y)
{
    const int lane = threadIdx.x & 31;
    const int wave = threadIdx.x >> 5;
    const int half = lane >> 4;
    const int l15  = lane & 15;

    const int t0 = (blockIdx.x * 4 + wave) * 16;
    const int hh = blockIdx.y;
    const int b  = blockIdx.z;

    const size_t base = (size_t)b * Tn * Cn + (size_t)hh * Dn;
    const float* Qp = q + base;
    const float* Kp = k + base;
    const float* Vp = v + base;

    __shared__ float Pt[4][16][17];   // +1 pad: conflict-free transposed reads

    // Q fragments for all 16 K-steps over D=64 (resident for whole tile)
    v2f aq[16];
    {
        const float* qrow = Qp + (size_t)(t0 + l15) * Cn;
#pragma unroll
        for (int s = 0; s < 16; ++s) {
            const int dd = s * 4 + half * 2;
            aq[s].x = qrow[dd];
            aq[s].y = qrow[dd + 1];
        }
    }

    v8f o[4] = {};
    float mrow[8], lrow[8];
#pragma unroll
    for (int r = 0; r < 8; ++r) { mrow[r] = -1e30f; lrow[r] = 0.0f; }

    for (int s0 = 0; s0 <= t0; s0 += 16) {
        // S = Q @ K^T  (16x16 tile, K-dim = 64)
        v8f sa = {};
#pragma unroll
        for (int s = 0; s < 16; ++s) {
            const int dd = s * 4 + half * 2;
            const float* kr = Kp + (size_t)(s0 + l15) * Cn;
            v2f bf;
            bf.x = kr[dd];
            bf.y = kr[dd + 1];
            sa = __builtin_amdgcn_wmma_f32_16x16x4_f32(
                false, aq[s], false, bf, 0, sa, false, false);
        }

        // scale, causal mask, online softmax update, write P to LDS
#pragma unroll
        for (int r = 0; r < 8; ++r) {
            const int row = t0 + r + half * 8;   // query index
            const int col = s0 + l15;            // key index
            float sv = sa[r] * 0.125f;           // 1/sqrt(64)
            if (col > row) sv = -1e30f;

            float mx = sv;
            mx = fmaxf(mx, __shfl_xor(mx, 1, 16));
            mx = fmaxf(mx, __shfl_xor(mx, 2, 16));
            mx = fmaxf(mx, __shfl_xor(mx, 4, 16));
            mx = fmaxf(mx, __shfl_xor(mx, 8, 16));

            const float mn    = fmaxf(mrow[r], mx);
            const float alpha = __expf(mrow[r] - mn);
            const float pv    = __expf(sv - mn);

            float ps = pv;
            ps += __shfl_xor(ps, 1, 16);
            ps += __shfl_xor(ps, 2, 16);
            ps += __shfl_xor(ps, 4, 16);
            ps += __shfl_xor(ps, 8, 16);

            lrow[r] = lrow[r] * alpha + ps;
            mrow[r] = mn;
#pragma unroll
            for (int j = 0; j < 4; ++j) o[j][r] *= alpha;

            Pt[wave][r + half * 8][l15] = pv;
        }
        // Same-wave LDS ops are in-order (DScnt); no cross-wave sharing here.

        // O += P @ V   (K-dim = 16 keys, N = 64 head dims)
#pragma unroll
        for (int ks = 0; ks < 16; ks += 4) {
            v2f ap;
            ap.x = Pt[wave][l15][ks + half * 2];
            ap.y = Pt[wave][l15][ks + half * 2 + 1];
#pragma unroll
            for (int j = 0; j < 4; ++j) {
                const int dd = j * 16 + l15;
                v2f bf;
                bf.x = Vp[(size_t)(s0 + ks + half * 2) * Cn + dd];
                bf.y = Vp[(size_t)(s0 + ks + half * 2 + 1) * Cn + dd];
                o[j] = __builtin_amdgcn_wmma_f32_16x16x4_f32(
                    false, ap, false, bf, 0, o[j], false, false);
            }
        }
    }

    // normalize and write y[b, t, h*64 + d]
#pragma unroll
    for (int j = 0; j < 4; ++j) {
        const int dd = j * 16 + l15;
#pragma unroll
        for (int r = 0; r < 8; ++r) {
            const int row = t0 + r + half * 8;
            y[base + (size_t)row * Cn + dd] = o[j][r] / lrow[r];
        }
    }
}

// ---------------------------------------------------------------------------
// LayerNorm: one 256-thread block per row of C=1024.
// ---------------------------------------------------------------------------
__global__ __launch_bounds__(256) void layernorm_kernel(
    const float* __restrict__ x, const float* __restrict__ g,
    const float* __restrict__ bta, float* __restrict__ out)
{
    const int row = blockIdx.x;
    const int tid = threadIdx.x;
    const float* xr = x + (size_t)row * Cn;
    __shared__ float red[256];

    float s = 0.0f;
    for (int i = tid; i < Cn; i += 256) s += xr[i];
    red[tid] = s;
    __syncthreads();
    for (int off = 128; off > 0; off >>= 1) {
        if (tid < off) red[tid] += red[tid + off];
        __syncthreads();
    }
    const float mu = red[0] * (1.0f / Cn);
    __syncthreads();

    float s2 = 0.0f;
    for (int i = tid; i < Cn; i += 256) {
        const float d = xr[i] - mu;
        s2 += d * d;
    }
    red[tid] = s2;
    __syncthreads();
    for (int off = 128; off > 0; off >>= 1) {
        if (tid < off) red[tid] += red[tid + off];
        __syncthreads();
    }
    const float rstd = rsqrtf(red[0] * (1.0f / Cn) + 1e-5f);
    __syncthreads();

    for (int i = tid; i < Cn; i += 256)
        out[(size_t)row * Cn + i] = (xr[i] - mu) * rstd * g[i] + bta[i];
}

// ---------------------------------------------------------------------------
// Router: logits = x1[0,t,:] @ router[C,8]; top-2, softmax -> gates[T,8].
// One wave per token.
// ---------------------------------------------------------------------------
__global__ __launch_bounds__(32) void router_kernel(
    const float* __restrict__ x1, const float* __restrict__ Wr,
    float* __restrict__ gates)
{
    const int t    = blockIdx.x;
    const int lane = threadIdx.x;
    const float* xr = x1 + (size_t)t * Cn;   // batch 0

    float acc[En] = {};
    for (int c = lane; c < Cn; c += 32) {
        const float xv = xr[c];
#pragma unroll
        for (int e = 0; e < En; ++e) acc[e] += xv * Wr[(size_t)c * En + e];
    }
#pragma unroll
    for (int e = 0; e < En; ++e) {
        float s = acc[e];
        s += __shfl_xor(s, 1, 32);
        s += __shfl_xor(s, 2, 32);
        s += __shfl_xor(s, 4, 32);
        s += __shfl_xor(s, 8, 32);
        s += __shfl_xor(s, 16, 32);
        acc[e] = s;
    }
    // top-2 (lowest index wins ties, matching lax.top_k), softmax of the pair
    int   i0 = 0;
    float v0 = acc[0];
#pragma unroll
    for (int e = 1; e < En; ++e)
        if (acc[e] > v0) { v0 = acc[e]; i0 = e; }
    int   i1 = -1;
    float v1 = -3.4e38f;
#pragma unroll
    for (int e = 0; e < En; ++e)
        if (e != i0 && acc[e] > v1) { v1 = acc[e]; i1 = e; }
    const float e0  = 1.0f;                 // exp(v0 - v0)
    const float e1  = __expf(v1 - v0);
    const float inv = 1.0f / (e0 + e1);
    if (lane < En) {
        const float gv = (lane == i0) ? e0 * inv : (lane == i1 ? e1 * inv : 0.0f);
        gates[(size_t)t * En + lane] = gv;
    }
}

__global__ void zero_kernel(float* __restrict__ p, int n)
{
    const int i = blockIdx.x * blockDim.x + threadIdx.x;
    if (i < n) p[i] = 0.0f;
}

// ---------------------------------------------------------------------------
// Host-side orchestration.
// Workspace layout (floats), total ~135 MB:
//   h    : B*T*C      (LN1 out, then reused for attention output y)
//   qb   : B*T*C      (Q, then reused for x1 = x + attn proj)
//   kb   : B*T*C      (K, then reused for y_adapt accumulator)
//   vb   : B*T*C      (V, then reused for h2 = LN2(x1))
//   mlp1 : B*T*4C
//   down : B*T*BOT    (per-expert adapter hidden, reused across experts)
//   gates: T*E
// ---------------------------------------------------------------------------
extern "C" void kernel_launch(void* const* d_in, const int* in_sizes, int n_in,
                              void* d_out, int out_size, void* d_ws, size_t ws_size,
                              hipStream_t stream)
{
    (void)in_sizes; (void)n_in; (void)out_size; (void)ws_size;

    const float* x      = (const float*)d_in[0];
    const float* Wq     = (const float*)d_in[1];
    const float* bq     = (const float*)d_in[2];
    const float* Wk     = (const float*)d_in[3];
    const float* bk     = (const float*)d_in[4];
    const float* Wv     = (const float*)d_in[5];
    const float* bv     = (const float*)d_in[6];
    const float* Wo     = (const float*)d_in[7];
    const float* bo     = (const float*)d_in[8];
    const float* ln1_g  = (const float*)d_in[9];
    const float* ln1_b  = (const float*)d_in[10];
    const float* ln2_g  = (const float*)d_in[11];
    const float* ln2_b  = (const float*)d_in[12];
    const float* mlp_w1 = (const float*)d_in[13];
    const float* mlp_b1 = (const float*)d_in[14];
    const float* mlp_w2 = (const float*)d_in[15];
    const float* mlp_b2 = (const float*)d_in[16];
    const float* router = (const float*)d_in[17];
    const float* ad_w   = (const float*)d_in[18];
    const float* ad_b   = (const float*)d_in[19];
    const float* au_w   = (const float*)d_in[20];
    const float* au_b   = (const float*)d_in[21];

    float* out = (float*)d_out;

    const size_t NBTC = (size_t)BTn * Cn;        // 4M floats
    float* ws    = (float*)d_ws;
    float* h     = ws;
    float* qb    = h    + NBTC;
    float* kb    = qb   + NBTC;
    float* vb    = kb   + NBTC;
    float* mlp1  = vb   + NBTC;
    float* down  = mlp1 + (size_t)BTn * FFn;
    float* gates = down + (size_t)BTn * BOTn;

    const dim3 gemmCC(Cn / 64, BTn / 128);       // M=4096, N=1024
    const dim3 gemmFF(FFn / 64, BTn / 128);      // M=4096, N=4096
    const dim3 gemmDN(BOTn / 64, BTn / 128);     // M=4096, N=64

    // 1. h = LN1(x)
    layernorm_kernel<<<BTn, 256, 0, stream>>>(x, ln1_g, ln1_b, h);

    // 2. Q/K/V projections
    gemm_wmma<0, false, false, false><<<gemmCC, 256, GEMM_LDS_BYTES, stream>>>(
        h, Wq, bq, nullptr, nullptr, nullptr, 0, qb, BTn, Cn, Cn);
    gemm_wmma<0, false, false, false><<<gemmCC, 256, GEMM_LDS_BYTES, stream>>>(
        h, Wk, bk, nullptr, nullptr, nullptr, 0, kb, BTn, Cn, Cn);
    gemm_wmma<0, false, false, false><<<gemmCC, 256, GEMM_LDS_BYTES, stream>>>(
        h, Wv, bv, nullptr, nullptr, nullptr, 0, vb, BTn, Cn, Cn);

    // 3. causal attention -> y (reuse h)
    attn_kernel<<<dim3(Tn / 16 / 4, Hn, Bn), 128, 0, stream>>>(qb, kb, vb, h);

    // 4. x1 = x + y @ Wo + bo   (into qb)
    gemm_wmma<0, true, false, false><<<gemmCC, 256, GEMM_LDS_BYTES, stream>>>(
        h, Wo, bo, x, nullptr, nullptr, 0, qb, BTn, Cn, Cn);

    // 5. router gates from batch 0 of x1
    router_kernel<<<Tn, 32, 0, stream>>>(qb, router, gates);

    // 6. adapter MoE: y_adapt (kb) = sum_e gate[t,e]*0.1*(relu(x1@adw_e+adb_e)@auw_e+aub_e)
    zero_kernel<<<(int)(NBTC / 256), 256, 0, stream>>>(kb, (int)NBTC);
    for (int e = 0; e < En; ++e) {
        gemm_wmma<1, false, false, false><<<gemmDN, 256, GEMM_LDS_BYTES, stream>>>(
            qb, ad_w + (size_t)e * Cn * BOTn, ad_b + (size_t)e * BOTn,
            nullptr, nullptr, nullptr, 0, down, BTn, BOTn, Cn);
        gemm_wmma<0, false, false, true><<<gemmCC, 256, GEMM_LDS_BYTES, stream>>>(
            down, au_w + (size_t)e * BOTn * Cn, au_b + (size_t)e * Cn,
            nullptr, nullptr, gates, e, kb, BTn, Cn, BOTn);
    }

    // 7. h2 = LN2(x1)  (into vb)
    layernorm_kernel<<<BTn, 256, 0, stream>>>(qb, ln2_g, ln2_b, vb);

    // 8. mlp1 = gelu(h2 @ W1 + b1)
    gemm_wmma<2, false, false, false><<<gemmFF, 256, GEMM_LDS_BYTES, stream>>>(
        vb, mlp_w1, mlp_b1, nullptr, nullptr, nullptr, 0, mlp1, BTn, FFn, Cn);

    // 9. out = x1 + y_adapt + (mlp1 @ W2 + b2)
    gemm_wmma<0, true, true, false><<<gemmCC, 256, GEMM_LDS_BYTES, stream>>>(
        mlp1, mlp_w2, mlp_b2, qb, kb, nullptr, 0, out, BTn, Cn, FFn);
}